// MPN_47261820125208
// MI455X (gfx1250) — compile-verified
//
#include <hip/hip_runtime.h>

typedef __attribute__((ext_vector_type(2))) float v2f;
typedef __attribute__((ext_vector_type(8))) float v8f;

#define BN_EPS_F 1e-5f

// Prepared-constants layout inside workspace (floats):
//   0   : B-frags we_w1 (3 chunks x 64)
//   192 : B-frag  we_w2 (1 x 64)
//   256 : B-frags wv_w1 (2 x 64)
//   384 : B-frag  wv_w2 (1 x 64)
//   448 : B-frag  out_w1 (1 x 64)
//   512 : B-frag  out_w2 (1 x 64)
//   576 : we_b1 pad16   592 : we_b2 pad16   608 : wv_b1 pad16
//   624 : wv_b2 pad16   640 : out_b1 pad16  656 : out_b2 pad16
//   672 : we_g1 pad4    676 : we_be1 pad4
//   680 : wv_g1 pad4    684 : wv_be1 pad4
//   688 : out_g pad4    692 : out_be pad4       (total 704)
#define P_SZ 704

// D = A(16x4,f32) * B(4x16,f32) + C, full-precision WMMA.
__device__ __forceinline__ v8f wmma4(v2f a, v2f b, v8f c) {
  return __builtin_amdgcn_wmma_f32_16x16x4_f32(false, a, false, b, (short)0, c,
                                               false, false);
}

// ---------------------------------------------------------------- zero ------
__global__ void k_zero(float* a, long na, float* b, long nb) {
  long i = (long)blockIdx.x * blockDim.x + threadIdx.x;
  if (i < na) a[i] = 0.f;
  if (i < nb) b[i] = 0.f;
}

// ------------------------------------------------------------- prep ---------
__device__ __forceinline__ void make_frag(float* dst, const float* w, int K,
                                          int Nw, int nchunks, int lane) {
  int hv = lane >> 4, col = lane & 15;
  for (int c = 0; c < nchunks; ++c) {
    int k0 = c * 4 + 2 * hv;
    dst[(c * 32 + lane) * 2 + 0] = (k0 < K && col < Nw) ? w[k0 * Nw + col] : 0.f;
    dst[(c * 32 + lane) * 2 + 1] =
        (k0 + 1 < K && col < Nw) ? w[(k0 + 1) * Nw + col] : 0.f;
  }
}
__device__ __forceinline__ void pad_copy(float* dst, const float* s, int n,
                                         int m, int lane) {
  if (lane < m) dst[lane] = (lane < n) ? s[lane] : 0.f;
}

__global__ void k_prep(const float* we_w1, const float* we_b1, const float* we_g1,
                       const float* we_be1, const float* we_w2, const float* we_b2,
                       const float* wv_w1, const float* wv_b1, const float* wv_g1,
                       const float* wv_be1, const float* wv_w2, const float* wv_b2,
                       const float* out_w1, const float* out_b1, const float* out_g,
                       const float* out_be, const float* out_w2, const float* out_b2,
                       float* P) {
  int lane = threadIdx.x;
  if (lane >= 32) return;
  make_frag(P + 0, we_w1, 9, 4, 3, lane);
  make_frag(P + 192, we_w2, 4, 3, 1, lane);
  make_frag(P + 256, wv_w1, 6, 3, 2, lane);
  make_frag(P + 384, wv_w2, 3, 3, 1, lane);
  make_frag(P + 448, out_w1, 3, 3, 1, lane);
  make_frag(P + 512, out_w2, 3, 2, 1, lane);
  pad_copy(P + 576, we_b1, 4, 16, lane);
  pad_copy(P + 592, we_b2, 3, 16, lane);
  pad_copy(P + 608, wv_b1, 3, 16, lane);
  pad_copy(P + 624, wv_b2, 3, 16, lane);
  pad_copy(P + 640, out_b1, 3, 16, lane);
  pad_copy(P + 656, out_b2, 2, 16, lane);
  pad_copy(P + 672, we_g1, 4, 4, lane);
  pad_copy(P + 676, we_be1, 4, 4, lane);
  pad_copy(P + 680, wv_g1, 3, 4, lane);
  pad_copy(P + 684, wv_be1, 3, 4, lane);
  pad_copy(P + 688, out_g, 3, 4, lane);
  pad_copy(P + 692, out_be, 3, 4, lane);
}

// ------------- H row-major [E,3] -> tile-major [tile][3][16] ---------------
__global__ void k_h2t(const float* __restrict__ Hrow, float* __restrict__ Ht,
                      int E) {
  int i = blockIdx.x * blockDim.x + threadIdx.x;
  if (i < E * 3) {
    int e = i / 3, k = i - e * 3;
    Ht[(size_t)(e >> 4) * 48 + k * 16 + (e & 15)] = Hrow[i];
  }
}

// ------------- pass 1: t1 = e_in @ we_w1 + we_b1 ; column sum/sumsq --------
__global__ void k_edge1(const float* __restrict__ M, const float* __restrict__ Ht,
                        const int* __restrict__ ei, const float* __restrict__ P,
                        float* __restrict__ t1, float* __restrict__ stats, int E) {
  __shared__ float sred[8];
  int tid = threadIdx.x;
  if (tid < 8) sred[tid] = 0.f;
  __syncthreads();
  int lane = tid & 31, wv = tid >> 5, hv = lane >> 4, col = lane & 15;
  int tile = blockIdx.x * 8 + wv;
  int e = tile * 16 + col;
  bool valid = (e < E);
  int ec = valid ? e : 0;
  int si = ei[ec];
  int di = ei[E + ec];
  // unconditional gathers (clamped indices are in-bounds)
  float mi0 = M[(size_t)di * 3 + 0];
  float mi1 = M[(size_t)di * 3 + 1];
  float mi2 = M[(size_t)di * 3 + 2];
  float mj0 = M[(size_t)si * 3 + 0];
  float mj1 = M[(size_t)si * 3 + 1];
  float mj2 = M[(size_t)si * 3 + 2];
  size_t hb = (size_t)tile * 48 + col;
  float h0 = Ht[hb], h1 = Ht[hb + 16], h2 = Ht[hb + 32];
  const v2f* B = (const v2f*)P;
  v8f d = {0.f, 0.f, 0.f, 0.f, 0.f, 0.f, 0.f, 0.f};
  v2f a;
  // K slice (0,1)|(2,3): mi0,mi1 | mi2,mj0
  a.x = valid ? (hv ? mi2 : mi0) : 0.f;
  a.y = valid ? (hv ? mj0 : mi1) : 0.f;
  d = wmma4(a, B[lane], d);
  // K slice (4,5)|(6,7): mj1,mj2 | h0,h1
  a.x = valid ? (hv ? h0 : mj1) : 0.f;
  a.y = valid ? (hv ? h1 : mj2) : 0.f;
  d = wmma4(a, B[32 + lane], d);
  // K slice (8,9)|(10,11): h2,0 | 0,0
  a.x = valid ? (hv ? 0.f : h2) : 0.f;
  a.y = 0.f;
  d = wmma4(a, B[64 + lane], d);
  if (col < 4) {
    float bias = P[576 + col];
    float* base = t1 + (size_t)tile * 64 + col * 16 + 8 * hv;
    float s = 0.f, s2 = 0.f;
    if (tile * 16 + 16 <= E) {
      float4 lo = make_float4(d[0] + bias, d[1] + bias, d[2] + bias, d[3] + bias);
      float4 hi = make_float4(d[4] + bias, d[5] + bias, d[6] + bias, d[7] + bias);
      *(float4*)(base) = lo;
      *(float4*)(base + 4) = hi;
      s = lo.x + lo.y + lo.z + lo.w + hi.x + hi.y + hi.z + hi.w;
      s2 = lo.x * lo.x + lo.y * lo.y + lo.z * lo.z + lo.w * lo.w +
           hi.x * hi.x + hi.y * hi.y + hi.z * hi.z + hi.w * hi.w;
    } else {
#pragma unroll
      for (int v = 0; v < 8; ++v) {
        int er = tile * 16 + v + 8 * hv;
        float val = d[v] + bias;
        if (er < E) { base[v] = val; s += val; s2 += val * val; }
      }
    }
    atomicAdd(&sred[col], s);
    atomicAdd(&sred[4 + col], s2);
  }
  __syncthreads();
  if (tid < 8) atomicAdd(&stats[tid], sred[tid]);
}

// ---- pass 2: Hn = relu(bn(t1))@we_w2+b2 ; t2 = [Mi|Hn]@wv_w1+bv1 ; stats ---
__global__ void k_edge2(const float* __restrict__ M, const int* __restrict__ ei,
                        const float* __restrict__ t1, const float* __restrict__ P,
                        const float* __restrict__ stE, float* __restrict__ stV,
                        float* __restrict__ Hout, float* __restrict__ t2, int E) {
  __shared__ float hn[8][16][3];
  __shared__ float sred[8];
  int tid = threadIdx.x;
  if (tid < 8) sred[tid] = 0.f;
  int lane = tid & 31, wv = tid >> 5, hv = lane >> 4, col = lane & 15;
  int tile = blockIdx.x * 8 + wv;
  int e = tile * 16 + col;
  bool valid = (e < E);
  int ec = valid ? e : 0;
  bool full = (tile * 16 + 16 <= E);
  float invE = 1.f / (float)E;
  int k0 = 2 * hv, k1 = k0 + 1;
  float m0 = stE[k0] * invE, q0 = stE[4 + k0] * invE - m0 * m0;
  float m1 = stE[k1] * invE, q1 = stE[4 + k1] * invE - m1 * m1;
  float sc0 = P[672 + k0] * rsqrtf(q0 + BN_EPS_F), sh0 = P[676 + k0] - m0 * sc0;
  float sc1 = P[672 + k1] * rsqrtf(q1 + BN_EPS_F), sh1 = P[676 + k1] - m1 * sc1;
  size_t t1b = (size_t)tile * 64 + col;
  float x0 = t1[t1b + k0 * 16];
  float x1 = t1[t1b + k1 * 16];
  v2f a;
  a.x = valid ? fmaxf(x0 * sc0 + sh0, 0.f) : 0.f;
  a.y = valid ? fmaxf(x1 * sc1 + sh1, 0.f) : 0.f;
  v8f d = {0.f, 0.f, 0.f, 0.f, 0.f, 0.f, 0.f, 0.f};
  d = wmma4(a, ((const v2f*)(P + 192))[lane], d);
  if (col < 3) {
    float bias = P[592 + col];
    float4 lo = make_float4(d[0] + bias, d[1] + bias, d[2] + bias, d[3] + bias);
    float4 hi = make_float4(d[4] + bias, d[5] + bias, d[6] + bias, d[7] + bias);
    int r0 = 8 * hv;
    hn[wv][r0 + 0][col] = lo.x; hn[wv][r0 + 1][col] = lo.y;
    hn[wv][r0 + 2][col] = lo.z; hn[wv][r0 + 3][col] = lo.w;
    hn[wv][r0 + 4][col] = hi.x; hn[wv][r0 + 5][col] = hi.y;
    hn[wv][r0 + 6][col] = hi.z; hn[wv][r0 + 7][col] = hi.w;
    float* hbase = Hout + (size_t)tile * 48 + col * 16 + r0;
    if (full) {
      *(float4*)(hbase) = lo;
      *(float4*)(hbase + 4) = hi;
    } else {
      float va[8] = {lo.x, lo.y, lo.z, lo.w, hi.x, hi.y, hi.z, hi.w};
#pragma unroll
      for (int v = 0; v < 8; ++v)
        if (tile * 16 + r0 + v < E) hbase[v] = va[v];
    }
  }
  __syncthreads();
  // v_in = [Mi(3) | Hn(3)], K padded to 8 (two K=4 slices)
  int di = ei[E + ec];
  float mi0 = M[(size_t)di * 3 + 0];
  float mi1 = M[(size_t)di * 3 + 1];
  float mi2 = M[(size_t)di * 3 + 2];
  float g0 = hn[wv][col][0], g1 = hn[wv][col][1], g2 = hn[wv][col][2];
  v8f d2 = {0.f, 0.f, 0.f, 0.f, 0.f, 0.f, 0.f, 0.f};
  // K slice (0,1)|(2,3): mi0,mi1 | mi2,hn0
  a.x = valid ? (hv ? mi2 : mi0) : 0.f;
  a.y = valid ? (hv ? g0 : mi1) : 0.f;
  d2 = wmma4(a, ((const v2f*)(P + 256))[lane], d2);
  // K slice (4,5)|(6,7): hn1,hn2 | 0,0
  a.x = valid ? (hv ? 0.f : g1) : 0.f;
  a.y = valid ? (hv ? 0.f : g2) : 0.f;
  d2 = wmma4(a, ((const v2f*)(P + 256))[32 + lane], d2);
  if (col < 3) {
    float bias = P[608 + col];
    float* base = t2 + (size_t)tile * 48 + col * 16 + 8 * hv;
    float s = 0.f, s2 = 0.f;
    if (full) {
      float4 lo = make_float4(d2[0] + bias, d2[1] + bias, d2[2] + bias, d2[3] + bias);
      float4 hi = make_float4(d2[4] + bias, d2[5] + bias, d2[6] + bias, d2[7] + bias);
      *(float4*)(base) = lo;
      *(float4*)(base + 4) = hi;
      s = lo.x + lo.y + lo.z + lo.w + hi.x + hi.y + hi.z + hi.w;
      s2 = lo.x * lo.x + lo.y * lo.y + lo.z * lo.z + lo.w * lo.w +
           hi.x * hi.x + hi.y * hi.y + hi.z * hi.z + hi.w * hi.w;
    } else {
#pragma unroll
      for (int v = 0; v < 8; ++v) {
        int er = tile * 16 + v + 8 * hv;
        float val = d2[v] + bias;
        if (er < E) { base[v] = val; s += val; s2 += val * val; }
      }
    }
    atomicAdd(&sred[col], s);
    atomicAdd(&sred[4 + col], s2);
  }
  __syncthreads();
  if (tid < 8) atomicAdd(&stV[tid], sred[tid]);
}

// ---- pass 3: Mmsg = relu(bn(t2))@wv_w2+b2 ; atomic scatter-add to Mout ----
__global__ void k_edge3(const int* __restrict__ ei, const float* __restrict__ t2,
                        const float* __restrict__ P, const float* __restrict__ stV,
                        float* __restrict__ Mout, int E) {
  int tid = threadIdx.x;
  int lane = tid & 31, wv = tid >> 5, hv = lane >> 4, col = lane & 15;
  int tile = blockIdx.x * 8 + wv;
  int e = tile * 16 + col;
  bool valid = (e < E);
  float invE = 1.f / (float)E;
  int k0 = 2 * hv, k1 = k0 + 1;   // k1 may be the K=3 pad slot (params padded to 0)
  float m0 = stV[k0] * invE, q0 = stV[4 + k0] * invE - m0 * m0;
  float m1 = stV[k1] * invE, q1 = stV[4 + k1] * invE - m1 * m1;
  float sc0 = P[680 + k0] * rsqrtf(q0 + BN_EPS_F), sh0 = P[684 + k0] - m0 * sc0;
  float sc1 = P[680 + k1] * rsqrtf(q1 + BN_EPS_F), sh1 = P[684 + k1] - m1 * sc1;
  size_t tb = (size_t)tile * 48 + col;
  float x0 = t2[tb + k0 * 16];
  float x1 = t2[tb + k1 * 16];    // k1==3 reads padded tail; masked below
  v2f a;
  a.x = valid ? fmaxf(x0 * sc0 + sh0, 0.f) : 0.f;
  a.y = (valid && k1 < 3) ? fmaxf(x1 * sc1 + sh1, 0.f) : 0.f;
  v8f d = {0.f, 0.f, 0.f, 0.f, 0.f, 0.f, 0.f, 0.f};
  d = wmma4(a, ((const v2f*)(P + 384))[lane], d);
  if (col < 3) {
    float bias = P[624 + col];
#pragma unroll
    for (int v = 0; v < 8; ++v) {
      int er = tile * 16 + v + 8 * hv;
      if (er < E) {
        int di = ei[E + er];
        atomicAdd(&Mout[(size_t)di * 3 + col], d[v] + bias);
      }
    }
  }
}

// ---------------- head pass 1: t3 = H@out_w1 + b1 ; stats -------------------
__global__ void k_head1(const float* __restrict__ Ht, const float* __restrict__ P,
                        float* __restrict__ t3, float* __restrict__ stats, int E) {
  __shared__ float sred[8];
  int tid = threadIdx.x;
  if (tid < 8) sred[tid] = 0.f;
  __syncthreads();
  int lane = tid & 31, wv = tid >> 5, hv = lane >> 4, col = lane & 15;
  int tile = blockIdx.x * 8 + wv;
  int e = tile * 16 + col;
  bool valid = (e < E);
  int k0 = 2 * hv, k1 = k0 + 1;
  size_t hb = (size_t)tile * 48 + col;
  float x0 = Ht[hb + k0 * 16];
  float x1 = Ht[hb + k1 * 16];  // k1==3 reads padded tail; masked below
  v2f a;
  a.x = valid ? x0 : 0.f;
  a.y = (valid && k1 < 3) ? x1 : 0.f;
  v8f d = {0.f, 0.f, 0.f, 0.f, 0.f, 0.f, 0.f, 0.f};
  d = wmma4(a, ((const v2f*)(P + 448))[lane], d);
  if (col < 3) {
    float bias = P[640 + col];
    float* base = t3 + (size_t)tile * 48 + col * 16 + 8 * hv;
    float s = 0.f, s2 = 0.f;
    if (tile * 16 + 16 <= E) {
      float4 lo = make_float4(d[0] + bias, d[1] + bias, d[2] + bias, d[3] + bias);
      float4 hi = make_float4(d[4] + bias, d[5] + bias, d[6] + bias, d[7] + bias);
      *(float4*)(base) = lo;
      *(float4*)(base + 4) = hi;
      s = lo.x + lo.y + lo.z + lo.w + hi.x + hi.y + hi.z + hi.w;
      s2 = lo.x * lo.x + lo.y * lo.y + lo.z * lo.z + lo.w * lo.w +
           hi.x * hi.x + hi.y * hi.y + hi.z * hi.z + hi.w * hi.w;
    } else {
#pragma unroll
      for (int v = 0; v < 8; ++v) {
        int er = tile * 16 + v + 8 * hv;
        float val = d[v] + bias;
        if (er < E) { base[v] = val; s += val; s2 += val * val; }
      }
    }
    atomicAdd(&sred[col], s);
    atomicAdd(&sred[4 + col], s2);
  }
  __syncthreads();
  if (tid < 8) atomicAdd(&stats[tid], sred[tid]);
}

// -------- head pass 2: logits = relu(bn(t3))@out_w2 + b2 ; softmax ----------
__global__ void k_head2(const float* __restrict__ t3, const float* __restrict__ P,
                        const float* __restrict__ stH, float* __restrict__ out,
                        int E) {
  __shared__ float lg[8][16][2];
  int tid = threadIdx.x;
  int lane = tid & 31, wv = tid >> 5, hv = lane >> 4, col = lane & 15;
  int tile = blockIdx.x * 8 + wv;
  int e = tile * 16 + col;
  bool valid = (e < E);
  float invE = 1.f / (float)E;
  int k0 = 2 * hv, k1 = k0 + 1;
  float m0 = stH[k0] * invE, q0 = stH[4 + k0] * invE - m0 * m0;
  float m1 = stH[k1] * invE, q1 = stH[4 + k1] * invE - m1 * m1;
  float sc0 = P[688 + k0] * rsqrtf(q0 + BN_EPS_F), sh0 = P[692 + k0] - m0 * sc0;
  float sc1 = P[688 + k1] * rsqrtf(q1 + BN_EPS_F), sh1 = P[692 + k1] - m1 * sc1;
  size_t tb = (size_t)tile * 48 + col;
  float x0 = t3[tb + k0 * 16];
  float x1 = t3[tb + k1 * 16];
  v2f a;
  a.x = valid ? fmaxf(x0 * sc0 + sh0, 0.f) : 0.f;
  a.y = (valid && k1 < 3) ? fmaxf(x1 * sc1 + sh1, 0.f) : 0.f;
  v8f d = {0.f, 0.f, 0.f, 0.f, 0.f, 0.f, 0.f, 0.f};
  d = wmma4(a, ((const v2f*)(P + 512))[lane], d);
  if (col < 2) {
    float bias = P[656 + col];
#pragma unroll
    for (int v = 0; v < 8; ++v) lg[wv][v + 8 * hv][col] = d[v] + bias;
  }
  __syncthreads();
  if (col < 2) {
#pragma unroll
    for (int v = 0; v < 8; ++v) {
      int r = v + 8 * hv;
      int er = tile * 16 + r;
      if (er < E) {
        float l0 = lg[wv][r][0], l1 = lg[wv][r][1];
        float mx = fmaxf(l0, l1);
        float e0 = __expf(l0 - mx), e1 = __expf(l1 - mx);
        out[(size_t)er * 2 + col] = ((col == 0) ? e0 : e1) / (e0 + e1);
      }
    }
  }
}

// ---------------------------------------------------------------------------
extern "C" void kernel_launch(void* const* d_in, const int* in_sizes, int n_in,
                              void* d_out, int out_size, void* d_ws, size_t ws_size,
                              hipStream_t stream) {
  (void)n_in; (void)out_size; (void)ws_size;
  const float* M0 = (const float*)d_in[0];
  const float* H0 = (const float*)d_in[1];
  const int* ei = (const int*)d_in[2];

  int N = in_sizes[0] / 3;
  int E = in_sizes[1] / 3;
  size_t ET = (size_t)((E + 15) / 16);

  float* ws = (float*)d_ws;
  float* P = ws;                              // 704 (multiple of 16)
  float* stats = ws + P_SZ;                   // 24 used, reserve 32 -> next off 736
  float* Mb0 = ws + P_SZ + 32;
  float* Mb1 = Mb0 + (size_t)N * 3;
  float* Hb0 = Mb1 + (size_t)N * 3;
  float* Hb1 = Hb0 + ET * 48 + 64;
  float* t1 = Hb1 + ET * 48 + 64;             // [ET][4][16] + pad (also t3)
  float* t2 = t1 + ET * 64 + 64;              // [ET][3][16] + pad

  int egrid = (E + 127) / 128;                // 8 waves/block, 16 edges/wave
  k_prep<<<1, 32, 0, stream>>>(
      (const float*)d_in[3], (const float*)d_in[4], (const float*)d_in[5],
      (const float*)d_in[6], (const float*)d_in[7], (const float*)d_in[8],
      (const float*)d_in[9], (const float*)d_in[10], (const float*)d_in[11],
      (const float*)d_in[12], (const float*)d_in[13], (const float*)d_in[14],
      (const float*)d_in[15], (const float*)d_in[16], (const float*)d_in[17],
      (const float*)d_in[18], (const float*)d_in[19], (const float*)d_in[20], P);
  k_h2t<<<(E * 3 + 255) / 256, 256, 0, stream>>>(H0, Hb1, E);

  const float* Min = M0;
  const float* Hin = Hb1;
  for (int l = 0; l < 20; ++l) {
    float* Mout = (l & 1) ? Mb1 : Mb0;
    float* Hout = (l & 1) ? Hb1 : Hb0;
    long zn = (long)N * 3;
    int zgrid = (int)((zn + 255) / 256);
    k_zero<<<zgrid, 256, 0, stream>>>(Mout, zn, stats, 16);
    k_edge1<<<egrid, 256, 0, stream>>>(Min, Hin, ei, P, t1, stats, E);
    k_edge2<<<egrid, 256, 0, stream>>>(Min, ei, t1, P, stats, stats + 8, Hout, t2, E);
    k_edge3<<<egrid, 256, 0, stream>>>(ei, t2, P, stats + 8, Mout, E);
    Min = Mout;
    Hin = Hout;
  }
  k_zero<<<1, 256, 0, stream>>>(stats + 16, 8, nullptr, 0);
  k_head1<<<egrid, 256, 0, stream>>>(Hin, P, t1, stats + 16, E);
  k_head2<<<egrid, 256, 0, stream>>>(t1, P, stats + 16, (float*)d_out, E);
}